// InfoNCE_35673998361054
// MI455X (gfx1250) — compile-verified
//
#include <hip/hip_runtime.h>
#include <math.h>

typedef __attribute__((ext_vector_type(2))) float v2f;
typedef __attribute__((ext_vector_type(8))) float v8f;

#define B_SZ 256
#define C_SZ 256
#define S_SZ 256   // 16*16 spatial elements per (b,c)
#define R_SZ 512   // 2*B rows of rep

__device__ __forceinline__ float wave_reduce_sum(float v) {
#pragma unroll
    for (int off = 16; off > 0; off >>= 1)
        v += __shfl_xor(v, off, 32);
    return v;
}

// ---------------------------------------------------------------------------
// Kernel 1: spatial means. One wave32 per (row, channel): 256 contiguous
// floats -> 2x float4 per lane (global_load_b128), shfl-xor wave reduce.
// rep rows [0,256) = mean(zjs), rows [256,512) = mean(zis)  (rep = [zj; zi]).
// This kernel is the bandwidth-bound part: 128 MB streamed.
// ---------------------------------------------------------------------------
__global__ void mean_kernel(const float* __restrict__ zis,
                            const float* __restrict__ zjs,
                            float* __restrict__ rep) {
    int wave = (int)((blockIdx.x * blockDim.x + threadIdx.x) >> 5);
    int lane = threadIdx.x & 31;
    int r = wave >> 8;    // 0..511
    int c = wave & 255;   // 0..255
    const float* src = (r < B_SZ)
        ? (zjs + ((size_t)r * C_SZ + c) * S_SZ)
        : (zis + ((size_t)(r - B_SZ) * C_SZ + c) * S_SZ);
    float4 a = ((const float4*)src)[lane];
    float4 b = ((const float4*)src)[lane + 32];
    float s = (a.x + a.y) + (a.z + a.w) + (b.x + b.y) + (b.z + b.w);
    s = wave_reduce_sum(s);
    if (lane == 0) rep[wave] = s * (1.0f / S_SZ);
}

// ---------------------------------------------------------------------------
// Kernel 2: L2-normalize each of the 512 rows (one wave per row).
// ---------------------------------------------------------------------------
__global__ void norm_kernel(float* __restrict__ rep) {
    int wave = (int)((blockIdx.x * blockDim.x + threadIdx.x) >> 5);  // 0..511
    int lane = threadIdx.x & 31;
    float* row = rep + (size_t)wave * C_SZ;
    float v[8];
    float s = 0.0f;
#pragma unroll
    for (int i = 0; i < 8; i++) {
        v[i] = row[lane + 32 * i];
        s += v[i] * v[i];
    }
    s = wave_reduce_sum(s);
    float nrm = fmaxf(sqrtf(s), 1e-12f);
    float inv = 1.0f / nrm;
#pragma unroll
    for (int i = 0; i < 8; i++) row[lane + 32 * i] = v[i] * inv;
}

// ---------------------------------------------------------------------------
// Kernel 3: sim = repn @ repn^T via V_WMMA_F32_16X16X4_F32 (full fp32 path).
// One wave per 16x16 output tile; K=256 -> 64 WMMA steps.
// A (16x4 f32) layout: lanes 0-15 M=0..15 K={k,k+1}; lanes 16-31 K={k+2,k+3}.
// B (4x16 f32) layout mirrors it with N=lane&15, so for N*N^T both fragments
// are the same float2 load from the normalized rep matrix.
// ---------------------------------------------------------------------------
__global__ void simgemm_kernel(const float* __restrict__ repn,
                               float* __restrict__ sim) {
    int wave = (int)((blockIdx.x * blockDim.x + threadIdx.x) >> 5);  // 0..1023
    int lane = threadIdx.x & 31;
    int tm = wave >> 5;   // 0..31 (512/16 tile rows)
    int tn = wave & 31;   // 0..31 tile cols
    int idx  = lane & 15;
    int koff = (lane >> 4) << 1;  // 0 or 2
    const float* arow = repn + (size_t)(tm * 16 + idx) * C_SZ + koff;
    const float* brow = repn + (size_t)(tn * 16 + idx) * C_SZ + koff;
    v8f acc = {};
#pragma unroll 8
    for (int k0 = 0; k0 < C_SZ; k0 += 4) {
        v2f a = *(const v2f*)(arow + k0);
        v2f b = *(const v2f*)(brow + k0);
        acc = __builtin_amdgcn_wmma_f32_16x16x4_f32(
            /*neg_a=*/false, a, /*neg_b=*/false, b,
            /*c_mod=*/(short)0, acc, /*reuse_a=*/false, /*reuse_b=*/false);
    }
    // C/D layout: VGPR i -> row tm*16 + i (+8 for lanes 16-31), col = tn*16 + lane&15
    int rbase = tm * 16 + ((lane >> 4) << 3);
    int col   = tn * 16 + idx;
#pragma unroll
    for (int i = 0; i < 8; i++)
        sim[(size_t)(rbase + i) * R_SZ + col] = acc[i];
}

// ---------------------------------------------------------------------------
// Kernel 4: cross-block row sums of exp(sim).
// srow[r] = sum_{c in other block} exp(sim[r][c]); one wave per row.
// ---------------------------------------------------------------------------
__global__ void srow_kernel(const float* __restrict__ sim,
                            float* __restrict__ srow) {
    int wave = (int)((blockIdx.x * blockDim.x + threadIdx.x) >> 5);  // 0..511
    int lane = threadIdx.x & 31;
    int cbase = (wave < B_SZ) ? B_SZ : 0;
    const float* row = sim + (size_t)wave * R_SZ + cbase;
    float s = 0.0f;
#pragma unroll
    for (int i = 0; i < 8; i++) s += expf(row[lane + 32 * i]);
    s = wave_reduce_sum(s);
    if (lane == 0) srow[wave] = s;
}

// ---------------------------------------------------------------------------
// Kernel 5: final scalar. logit = sim - log(exp(sim) + srow[r]) over the two
// diagonal blocks; answer = (sumA + sumD) / (B*B). Single block, LDS tree
// reduction -> deterministic.
// ---------------------------------------------------------------------------
__global__ void final_kernel(const float* __restrict__ sim,
                             const float* __restrict__ srow,
                             float* __restrict__ out) {
    __shared__ float red[256];
    int tid = threadIdx.x;
    float acc = 0.0f;
    for (int idx = tid; idx < 2 * B_SZ * B_SZ; idx += 256) {
        int blk = idx >> 16;            // 0 = A block, 1 = D block
        int loc = idx & 0xFFFF;
        int r = (loc >> 8) + blk * B_SZ;
        int c = (loc & 255) + blk * B_SZ;
        float sv = sim[(size_t)r * R_SZ + c];
        acc += sv - logf(expf(sv) + srow[r]);
    }
    red[tid] = acc;
    __syncthreads();
#pragma unroll
    for (int s = 128; s > 0; s >>= 1) {
        if (tid < s) red[tid] += red[tid + s];
        __syncthreads();
    }
    if (tid == 0) out[0] = red[0] * (1.0f / (float)(B_SZ * B_SZ));
}

// ---------------------------------------------------------------------------
extern "C" void kernel_launch(void* const* d_in, const int* in_sizes, int n_in,
                              void* d_out, int out_size, void* d_ws, size_t ws_size,
                              hipStream_t stream) {
    (void)in_sizes; (void)n_in; (void)out_size; (void)ws_size;
    const float* zis = (const float*)d_in[0];
    const float* zjs = (const float*)d_in[1];
    float* rep  = (float*)d_ws;            // 512*256 floats (normalized in-place)
    float* sim  = rep + R_SZ * C_SZ;       // 512*512 floats
    float* srow = sim + R_SZ * R_SZ;       // 512 floats
    float* out  = (float*)d_out;

    // 131072 waves (one per (row, channel) reduction): 16384 blocks x 8 waves
    mean_kernel<<<16384, 256, 0, stream>>>(zis, zjs, rep);
    // 512 waves
    norm_kernel<<<64, 256, 0, stream>>>(rep);
    // 1024 tiles, one wave each: 256 blocks x 4 waves
    simgemm_kernel<<<256, 128, 0, stream>>>(rep, sim);
    // 512 waves
    srow_kernel<<<64, 256, 0, stream>>>(sim, srow);
    // single-block deterministic final reduction
    final_kernel<<<1, 256, 0, stream>>>(sim, srow, out);
}